// FPSWE_40303973105696
// MI455X (gfx1250) — compile-verified
//
#include <hip/hip_runtime.h>
#include <hip/hip_bf16.h>

typedef float v2f __attribute__((ext_vector_type(2)));
typedef float v4f __attribute__((ext_vector_type(4)));
typedef float v8f __attribute__((ext_vector_type(8)));

#define BB 8
#define NN 2048
#define DD 128
#define PP 1024
#define MM 1024

// ---------------------------------------------------------------------------
// Kernel 1: projection GEMM via V_WMMA_F32_16X16X4_F32.
// C[n, p] = sum_d X[b, n, d] * theta[p, d]   (theta row-major [P, D])
// One wave per 16x16 tile. Output stored transposed: Xp[b][p][n].
// ---------------------------------------------------------------------------
__global__ __launch_bounds__(256) void proj_wmma_kernel(
    const float* __restrict__ X, const float* __restrict__ Th,
    float* __restrict__ Xp) {
  const int lane = threadIdx.x & 31;
  const int wave = threadIdx.x >> 5;
  const int NT = NN / 16;            // 128 tiles along n
  const int PT = PP / 16;            // 64 tiles along p
  int tile = blockIdx.x * 8 + wave;  // 8 waves per block
  int b  = tile / (NT * PT);
  int r  = tile % (NT * PT);
  int n0 = (r / PT) * 16;
  int p0 = (r % PT) * 16;

  // A (16x4 f32): lanes 0-15 hold rows M=0..15; VGPR0 = K={0|2}, VGPR1 = K={1|3}
  // -> lane supplies float2 {A[m,kb], A[m,kb+1]}, kb = (lane>>4)*2
  // B (4x16 f32): row striped across lanes; lane supplies {B[kb,n], B[kb+1,n]}
  //   with B[k,n] = theta[p0+n, k].
  const int mrow = lane & 15;
  const int kb   = (lane >> 4) << 1;
  const float* Arow = X  + ((size_t)b * NN + n0 + mrow) * DD + kb;
  const float* Brow = Th + ((size_t)(p0 + mrow)) * DD + kb;

  v8f c = {};
  #pragma unroll
  for (int k = 0; k < DD; k += 4) {
    v2f a  = *(const v2f*)(Arow + k);
    v2f bm = *(const v2f*)(Brow + k);
    c = __builtin_amdgcn_wmma_f32_16x16x4_f32(
        /*neg_a=*/false, a, /*neg_b=*/false, bm,
        /*c_mod=*/(short)0, c, /*reuse_a=*/false, /*reuse_b=*/false);
  }

  // D layout: VGPR r holds C[r, lane] (lanes 0-15) / C[r+8, lane-16] (16-31).
  // m (= n index) is contiguous across r -> two 16B stores per lane.
  const int pcol  = lane & 15;
  const int nbase = n0 + ((lane >> 4) << 3);
  float* dst = Xp + ((size_t)b * PP + p0 + pcol) * NN + nbase;
  *(v4f*)(dst)     = v4f{c[0], c[1], c[2], c[3]};
  *(v4f*)(dst + 4) = v4f{c[4], c[5], c[6], c[7]};
}

// ---------------------------------------------------------------------------
// Kernel 2: per-(b,p) row: LDS bitonic sort of 2048 values, then fused
// quantile interpolation + weighted reduction:
//   out[b,p] = sum_m (ref[m] - Xi[m]) * W[p,m]
// ---------------------------------------------------------------------------
__global__ __launch_bounds__(256) void sort_interp_reduce_kernel(
    const float* __restrict__ Xp, const float* __restrict__ ref,
    const float* __restrict__ W, float* __restrict__ out) {
  __shared__ float s[NN];
  __shared__ float red[256];
  const int tid = threadIdx.x;
  const int bp  = blockIdx.x;     // b*P + p
  const int p   = bp & (PP - 1);

  const float* row = Xp + (size_t)bp * NN;
  for (int i = tid; i < NN; i += 256) s[i] = row[i];
  __syncthreads();

  // Bitonic sort, ascending.
  for (unsigned k = 2; k <= NN; k <<= 1) {
    for (unsigned j = k >> 1; j > 0; j >>= 1) {
      for (unsigned i = tid; i < NN; i += 256) {
        unsigned ixj = i ^ j;
        if (ixj > i) {
          float a = s[i], c = s[ixj];
          bool up = ((i & k) == 0);
          bool sw = up ? (a > c) : (a < c);
          if (sw) { s[i] = c; s[ixj] = a; }
        }
      }
      __syncthreads();
    }
  }

  // Quantile interp (grids: x1d(i)=(i+1)/2049, xnew(m)=(m+1)/1025) + reduce.
  const float invN1 = 1.0f / (float)(NN + 1);
  const float invM1 = 1.0f / (float)(MM + 1);
  const float eps   = 1.1920929e-7f;  // FLT_EPSILON
  float acc = 0.0f;
  for (int m = tid; m < MM; m += 256) {
    float xnew = (float)(m + 1) * invM1;
    // searchsorted(left): smallest i with x1d(i) >= xnew
    int lo = 0, hi = NN;
    while (lo < hi) {
      int mid = (lo + hi) >> 1;
      if ((float)(mid + 1) * invN1 < xnew) lo = mid + 1; else hi = mid;
    }
    int ind = lo - 1;
    if (ind < 0) ind = 0;
    if (ind > NN - 2) ind = NN - 2;
    float x0 = (float)(ind + 1) * invN1;
    float x1 = (float)(ind + 2) * invN1;
    float t  = xnew - x0;
    float xi = s[ind] + (s[ind + 1] - s[ind]) / (eps + (x1 - x0)) * t;
    acc += (ref[m] - xi) * W[(size_t)p * MM + m];
  }
  red[tid] = acc;
  __syncthreads();
  for (int off = 128; off > 0; off >>= 1) {
    if (tid < off) red[tid] += red[tid + off];
    __syncthreads();
  }
  if (tid == 0) out[bp] = red[0];
}

extern "C" void kernel_launch(void* const* d_in, const int* in_sizes, int n_in,
                              void* d_out, int out_size, void* d_ws, size_t ws_size,
                              hipStream_t stream) {
  const float* X     = (const float*)d_in[0];  // [B, N, D]
  const float* theta = (const float*)d_in[1];  // [P, D]
  const float* ref   = (const float*)d_in[2];  // [M]
  const float* W     = (const float*)d_in[3];  // [P, M]
  float* out = (float*)d_out;                  // [B, P]
  float* Xp  = (float*)d_ws;                   // [B, P, N]  64 MB

  const int tiles = BB * (NN / 16) * (PP / 16);  // 65536, 8 waves/block
  proj_wmma_kernel<<<tiles / 8, 256, 0, stream>>>(X, theta, Xp);
  sort_interp_reduce_kernel<<<BB * PP, 256, 0, stream>>>(Xp, ref, W, out);
}